// Model_33672543601281
// MI455X (gfx1250) — compile-verified
//
#include <hip/hip_runtime.h>

typedef float v2f __attribute__((ext_vector_type(2)));
typedef float v8f __attribute__((ext_vector_type(8)));

#define NEG_SLOPE 0.01f

__device__ __forceinline__ float lrelu(float x) {
    return x >= 0.0f ? x : NEG_SLOPE * x;
}

// D = A(16x4 f32) * B(4x16 f32) + C(16x16 f32), exact f32 WMMA on gfx1250
__device__ __forceinline__ v8f wmma4(v2f a, v2f b, v8f c) {
    return __builtin_amdgcn_wmma_f32_16x16x4_f32(false, a, false, b, (short)0, c,
                                                 false, false);
}

// float atomic max via sign-split trick (works for mixed signs, -inf init)
__device__ __forceinline__ void atomicMaxF(float* addr, float val) {
    if (val >= 0.0f) atomicMax((int*)addr, __float_as_int(val));
    else             atomicMin((unsigned int*)addr, (unsigned int)__float_as_int(val));
}

__global__ __launch_bounds__(256) void fill_kernel(float* __restrict__ p, float v, int n) {
    int i = blockIdx.x * 256 + threadIdx.x;
    if (i < n) p[i] = v;
}

// Y[M,64] = X[M,64] @ W[64,64] + b[64], one wave per 16-row slab, WMMA f32 16x16x4
__global__ __launch_bounds__(256) void gemm64_kernel(
    const float* __restrict__ X, const float* __restrict__ W,
    const float* __restrict__ b, float* __restrict__ Y, int M)
{
    __shared__ __align__(16) float sWt[64 * 66];  // transposed: sWt[col*66 + k]
    for (int i = threadIdx.x; i < 4096; i += 256) {
        int k = i >> 6, j = i & 63;
        sWt[j * 66 + k] = W[i];
    }
    __syncthreads();

    const int wave = threadIdx.x >> 5;
    const int lane = threadIdx.x & 31;
    const int n  = lane & 15;
    const int hi = lane >> 4;
    const int rowbase = blockIdx.x * 128 + wave * 16;

    int arow = rowbase + n;
    if (arow >= M) arow = M - 1;
    const float* xrow = X + (size_t)arow * 64;

    v8f acc[4];
#pragma unroll
    for (int nt = 0; nt < 4; ++nt) {
        float bc = b[nt * 16 + n];
#pragma unroll
        for (int v = 0; v < 8; ++v) acc[nt][v] = bc;
    }

#pragma unroll
    for (int ks = 0; ks < 16; ++ks) {
        const int k0 = ks * 4 + 2 * hi;
        v2f a = *(const v2f*)(xrow + k0);
#pragma unroll
        for (int nt = 0; nt < 4; ++nt) {
            v2f bf = *(const v2f*)(&sWt[(nt * 16 + n) * 66 + k0]);
            acc[nt] = wmma4(a, bf, acc[nt]);
        }
    }

#pragma unroll
    for (int nt = 0; nt < 4; ++nt) {
#pragma unroll
        for (int v = 0; v < 8; ++v) {
            int row = rowbase + v + 8 * hi;
            if (row < M) Y[(size_t)row * 64 + nt * 16 + n] = acc[nt][v];
        }
    }
}

// Fused per-edge kernel: 4 waves x 16 edges per block.
//   hd   = (0.5*(Xu[ru]+Xi[ri])) @ Wdst + bdst      (WMMA, registers only)
//   a_u/a_i attention -> h_rev (written to HBM + LDS staging)
//   h_o  = h_rev @ Wo + bo                          (WMMA, registers only)
//   aggregator logits for both sides + atomic segment-max
__global__ __launch_bounds__(128) void edge_kernel(
    const float* __restrict__ Xu, const float* __restrict__ Xi,
    const float* __restrict__ Wdst, const float* __restrict__ bdst,
    const float* __restrict__ HSu, const float* __restrict__ HSi,
    const float* __restrict__ watt_rep, const float* __restrict__ batt_rep,
    const float* __restrict__ Wo, const float* __restrict__ bo,
    const float* __restrict__ Pu, const float* __restrict__ Pi,
    const float* __restrict__ watt_agg, const float* __restrict__ batt_agg,
    const int* __restrict__ r_user, const int* __restrict__ r_item,
    float* __restrict__ hrev, float* __restrict__ aU, float* __restrict__ aI,
    float* __restrict__ mU, float* __restrict__ mI, int R)
{
    __shared__ __align__(16) float sWdst[64 * 66];
    __shared__ __align__(16) float sWo[64 * 66];
    __shared__ __align__(16) float sStage[4][16 * 66];

    for (int i = threadIdx.x; i < 4096; i += 128) {
        int k = i >> 6, j = i & 63;
        sWdst[j * 66 + k] = Wdst[i];
        sWo[j * 66 + k]   = Wo[i];
    }
    __syncthreads();

    const int wave = threadIdx.x >> 5;
    const int lane = threadIdx.x & 31;
    const int n  = lane & 15;
    const int hi = lane >> 4;
    const int ebase = blockIdx.x * 64 + wave * 16;

    // A-fragment source: this lane supplies row (n) of the 16-edge tile
    int eA = ebase + n; if (eA >= R) eA = R - 1;
    const float* xuA = Xu + (size_t)r_user[eA] * 64;
    const float* xiA = Xi + (size_t)r_item[eA] * 64;

    // ---- GEMM1: hd = agg @ Wdst + bdst ----
    v8f acc[4];
#pragma unroll
    for (int nt = 0; nt < 4; ++nt) {
        float bc = bdst[nt * 16 + n];
#pragma unroll
        for (int v = 0; v < 8; ++v) acc[nt][v] = bc;
    }
#pragma unroll
    for (int ks = 0; ks < 16; ++ks) {
        const int k0 = ks * 4 + 2 * hi;
        v2f xa = *(const v2f*)(xuA + k0);
        v2f xb = *(const v2f*)(xiA + k0);
        v2f a = 0.5f * (xa + xb);
#pragma unroll
        for (int nt = 0; nt < 4; ++nt) {
            v2f bf = *(const v2f*)(&sWdst[(nt * 16 + n) * 66 + k0]);
            acc[nt] = wmma4(a, bf, acc[nt]);
        }
    }

    // edges owned by this lane in C-layout: e = v + 8*hi
    int eh[8], ruh[8], rih[8];
#pragma unroll
    for (int v = 0; v < 8; ++v) {
        int e = ebase + v + 8 * hi;
        if (e >= R) e = R - 1;
        eh[v] = e; ruh[v] = r_user[e]; rih[v] = r_item[e];
    }

    // ---- review attention: a_u, a_i ----
    const float brep = batt_rep[0];
    float pu_[8], pi_[8];
#pragma unroll
    for (int v = 0; v < 8; ++v) { pu_[v] = 0.0f; pi_[v] = 0.0f; }
#pragma unroll
    for (int nt = 0; nt < 4; ++nt) {
        const int col = nt * 16 + n;
        const float wr = watt_rep[col];
#pragma unroll
        for (int v = 0; v < 8; ++v) {
            float hd = acc[nt][v];
            pu_[v] += lrelu(HSu[(size_t)ruh[v] * 64 + col] + hd) * wr;
            pi_[v] += lrelu(HSi[(size_t)rih[v] * 64 + col] + hd) * wr;
        }
    }
#pragma unroll
    for (int m = 1; m < 16; m <<= 1) {   // butterfly over the 16 column-lanes
#pragma unroll
        for (int v = 0; v < 8; ++v) {
            pu_[v] += __shfl_xor(pu_[v], m, 32);
            pi_[v] += __shfl_xor(pi_[v], m, 32);
        }
    }
    float wu[8], wi[8];
#pragma unroll
    for (int v = 0; v < 8; ++v) {
        float au = __expf(pu_[v] + brep);
        float ai = __expf(pi_[v] + brep);
        float inv = 1.0f / (au + ai);
        wu[v] = au * inv; wi[v] = ai * inv;
    }

    // ---- h_rev: to LDS staging (stride 66) + global ----
#pragma unroll
    for (int v = 0; v < 8; ++v) {
        const int erow = v + 8 * hi;
        const bool valid = (ebase + erow) < R;
        const size_t go  = (size_t)eh[v] * 64;
        const size_t xou = (size_t)ruh[v] * 64;
        const size_t xoi = (size_t)rih[v] * 64;
#pragma unroll
        for (int nt = 0; nt < 4; ++nt) {
            const int col = nt * 16 + n;
            float h = wu[v] * Xu[xou + col] + wi[v] * Xi[xoi + col];
            sStage[wave][erow * 66 + col] = h;
            if (valid) hrev[go + col] = h;
        }
    }
    __syncthreads();

    // ---- GEMM2: ho = h_rev @ Wo + bo (A from LDS staging) ----
    v8f acc2[4];
#pragma unroll
    for (int nt = 0; nt < 4; ++nt) {
        float bc = bo[nt * 16 + n];
#pragma unroll
        for (int v = 0; v < 8; ++v) acc2[nt][v] = bc;
    }
#pragma unroll
    for (int ks = 0; ks < 16; ++ks) {
        const int k0 = ks * 4 + 2 * hi;
        v2f a = *(const v2f*)(&sStage[wave][n * 66 + k0]);
#pragma unroll
        for (int nt = 0; nt < 4; ++nt) {
            v2f bf = *(const v2f*)(&sWo[(nt * 16 + n) * 66 + k0]);
            acc2[nt] = wmma4(a, bf, acc2[nt]);
        }
    }

    // ---- aggregator attention logits (both sides) ----
    const float bagg = batt_agg[0];
    float qu[8], qi[8];
#pragma unroll
    for (int v = 0; v < 8; ++v) { qu[v] = 0.0f; qi[v] = 0.0f; }
#pragma unroll
    for (int nt = 0; nt < 4; ++nt) {
        const int col = nt * 16 + n;
        const float wa = watt_agg[col];
#pragma unroll
        for (int v = 0; v < 8; ++v) {
            float ho = acc2[nt][v];
            qu[v] += lrelu(ho + Pi[(size_t)rih[v] * 64 + col]) * wa;  // user side: pref_item
            qi[v] += lrelu(ho + Pu[(size_t)ruh[v] * 64 + col]) * wa;  // item side: pref_user
        }
    }
#pragma unroll
    for (int m = 1; m < 16; m <<= 1) {
#pragma unroll
        for (int v = 0; v < 8; ++v) {
            qu[v] += __shfl_xor(qu[v], m, 32);
            qi[v] += __shfl_xor(qi[v], m, 32);
        }
    }
    if (n < 8) {
        // lane n handles edge (ebase + n + 8*hi); select with cndmask chain (no spill)
        float lu = 0.0f, li = 0.0f; int ru = 0, ri = 0;
#pragma unroll
        for (int v = 0; v < 8; ++v)
            if (v == n) { lu = qu[v]; li = qi[v]; ru = ruh[v]; ri = rih[v]; }
        const int e = ebase + n + 8 * hi;
        if (e < R) {
            lu += bagg; li += bagg;
            aU[e] = lu; aI[e] = li;
            atomicMaxF(mU + ru, lu);
            atomicMaxF(mI + ri, li);
        }
    }
}

// stable edge-softmax: e = exp(a - m[dst]); atomic segment sums
__global__ __launch_bounds__(256) void softmax_kernel(
    const int* __restrict__ r_user, const int* __restrict__ r_item,
    float* __restrict__ aU, float* __restrict__ aI,
    const float* __restrict__ mU, const float* __restrict__ mI,
    float* __restrict__ sU, float* __restrict__ sI, int R)
{
    int r = blockIdx.x * 256 + threadIdx.x;
    if (r >= R) return;
    int ru = r_user[r], ri = r_item[r];
    float eu = __expf(aU[r] - mU[ru]);
    float ei = __expf(aI[r] - mI[ri]);
    aU[r] = eu; aI[r] = ei;
    atomicAdd(sU + ru, eu);
    atomicAdd(sI + ri, ei);
}

// out[dst] += h_rev * w  for both sides; one thread per (edge, col)
__global__ __launch_bounds__(256) void scatter_kernel(
    const int* __restrict__ r_user, const int* __restrict__ r_item,
    const float* __restrict__ aU, const float* __restrict__ aI,
    const float* __restrict__ sU, const float* __restrict__ sI,
    const float* __restrict__ hrev, float* __restrict__ out, int R, int nu)
{
    int gid = blockIdx.x * 256 + threadIdx.x;
    int r = gid >> 6, c = gid & 63;
    if (r >= R) return;
    int ru = r_user[r], ri = r_item[r];
    float wu = aU[r] / sU[ru];
    float wi = aI[r] / sI[ri];
    float h = hrev[(size_t)r * 64 + c];
    atomicAdd(out + (size_t)ru * 64 + c, h * wu);
    atomicAdd(out + ((size_t)(nu + ri)) * 64 + c, h * wi);
}

extern "C" void kernel_launch(void* const* d_in, const int* in_sizes, int n_in,
                              void* d_out, int out_size, void* d_ws, size_t ws_size,
                              hipStream_t stream)
{
    const float* Xu        = (const float*)d_in[0];
    const float* Xi        = (const float*)d_in[1];
    const float* Wsrc      = (const float*)d_in[2];
    const float* bsrc      = (const float*)d_in[3];
    const float* Wdst      = (const float*)d_in[4];
    const float* bdst      = (const float*)d_in[5];
    const float* watt_rep  = (const float*)d_in[6];
    const float* batt_rep  = (const float*)d_in[7];
    const float* Wo        = (const float*)d_in[8];
    const float* bo        = (const float*)d_in[9];
    const float* Wu        = (const float*)d_in[10];
    const float* bu        = (const float*)d_in[11];
    const float* watt_agg  = (const float*)d_in[12];
    const float* batt_agg  = (const float*)d_in[13];
    const float* pref_user = (const float*)d_in[14];
    const float* pref_item = (const float*)d_in[15];
    const int*   r_user    = (const int*)d_in[16];
    const int*   r_item    = (const int*)d_in[17];

    const int nu = in_sizes[0] / 64;
    const int ni = in_sizes[1] / 64;
    const int R  = in_sizes[16];

    float* ws  = (float*)d_ws;
    float* HSu = ws;  ws += (size_t)nu * 64;
    float* HSi = ws;  ws += (size_t)ni * 64;
    float* Pu  = ws;  ws += (size_t)nu * 64;
    float* Pi  = ws;  ws += (size_t)ni * 64;
    float* hrv = ws;  ws += (size_t)R * 64;
    float* aU  = ws;  ws += R;
    float* aI  = ws;  ws += R;
    float* mU  = ws;  ws += nu;   // mU, mI contiguous
    float* mI  = ws;  ws += ni;
    float* sU  = ws;  ws += nu;   // sU, sI contiguous
    float* sI  = ws;  ws += ni;

    float* out = (float*)d_out;
    const int outN = (nu + ni) * 64;
    const float ninf = -__builtin_huge_valf();

    fill_kernel<<<(nu + ni + 255) / 256, 256, 0, stream>>>(mU, ninf, nu + ni);
    fill_kernel<<<(nu + ni + 255) / 256, 256, 0, stream>>>(sU, 0.0f, nu + ni);
    fill_kernel<<<(outN + 255) / 256, 256, 0, stream>>>(out, 0.0f, outN);

    // per-node GEMMs (hoisted: Pu/Pi = pref @ Wu + bu is per-node, not per-edge)
    gemm64_kernel<<<(nu + 127) / 128, 256, 0, stream>>>(Xu, Wsrc, bsrc, HSu, nu);
    gemm64_kernel<<<(ni + 127) / 128, 256, 0, stream>>>(Xi, Wsrc, bsrc, HSi, ni);
    gemm64_kernel<<<(nu + 127) / 128, 256, 0, stream>>>(pref_user, Wu, bu, Pu, nu);
    gemm64_kernel<<<(ni + 127) / 128, 256, 0, stream>>>(pref_item, Wu, bu, Pi, ni);

    // fused edge pass (agg, hd, h_o never touch HBM)
    edge_kernel<<<(R + 63) / 64, 128, 0, stream>>>(
        Xu, Xi, Wdst, bdst, HSu, HSi, watt_rep, batt_rep, Wo, bo,
        Pu, Pi, watt_agg, batt_agg, r_user, r_item, hrv, aU, aI, mU, mI, R);

    softmax_kernel<<<(R + 255) / 256, 256, 0, stream>>>(
        r_user, r_item, aU, aI, mU, mI, sU, sI, R);

    const long tot = (long)R * 64;
    scatter_kernel<<<(int)((tot + 255) / 256), 256, 0, stream>>>(
        r_user, r_item, aU, aI, sU, sI, hrv, out, R, nu);

    (void)n_in; (void)out_size; (void)ws_size;
}